// FraudDetectionModel_65481071397884
// MI455X (gfx1250) — compile-verified
//
#include <hip/hip_runtime.h>
#include <hip/hip_bf16.h>

typedef __attribute__((ext_vector_type(2))) float v2f;
typedef __attribute__((ext_vector_type(8))) float v8f;

#define LOG2E 1.4426950408889634f

// tanh(x) = 1 - 2/(exp2(2*log2e*x)+1); uses v_exp_f32 + v_rcp_f32
__device__ __forceinline__ float fast_tanh(float x) {
    float xx = x * (2.0f * LOG2E);
    xx = fminf(fmaxf(xx, -30.0f), 30.0f);   // saturate well past tanh precision
    float e = __builtin_amdgcn_exp2f(xx);
    return fmaf(-2.0f, __builtin_amdgcn_rcpf(e + 1.0f), 1.0f);
}

// ---------------------------------------------------------------------------
// Kernel 1: the sequential photonic chain. One thread per sample.
// All parameter loads are wave-uniform -> scalar loads, stay hot in L2.
// Writes h[i] (float2) and sq[i] = ||h_i||^2 to workspace.
// ---------------------------------------------------------------------------
__global__ void k_chain(const float* __restrict__ x,
                        const float* __restrict__ in_W,  const float* __restrict__ in_b,
                        const float* __restrict__ in_s,  const float* __restrict__ in_sh,
                        const float* __restrict__ hid_W, const float* __restrict__ hid_b,
                        const float* __restrict__ hid_s, const float* __restrict__ hid_sh,
                        const float* __restrict__ fin_W, const float* __restrict__ fin_b,
                        float2* __restrict__ hout, float* __restrict__ sqout,
                        int H, int N) {
    int i = blockIdx.x * blockDim.x + threadIdx.x;
    if (i >= N) return;
    float x0 = x[2 * i], x1 = x[2 * i + 1];

    // input layer: tanh(x @ W.T + b) * scale + shift
    float h0 = fmaf(fast_tanh(fmaf(in_W[0], x0, fmaf(in_W[1], x1, in_b[0]))), in_s[0], in_sh[0]);
    float h1 = fmaf(fast_tanh(fmaf(in_W[2], x0, fmaf(in_W[3], x1, in_b[1]))), in_s[1], in_sh[1]);

    for (int t = 0; t < H; ++t) {
        const float* W = hid_W + 4 * t;
        float w0 = W[0], w1 = W[1], w2 = W[2], w3 = W[3];
        float b0 = hid_b[2 * t], b1 = hid_b[2 * t + 1];
        float s0 = hid_s[2 * t], s1 = hid_s[2 * t + 1];
        float p0 = hid_sh[2 * t], p1 = hid_sh[2 * t + 1];
        float a0 = fmaf(w0, h0, fmaf(w1, h1, b0));
        float a1 = fmaf(w2, h0, fmaf(w3, h1, b1));
        h0 = fmaf(fast_tanh(a0), s0, p0);
        h1 = fmaf(fast_tanh(a1), s1, p1);
    }

    float f0 = fmaf(fin_W[0], h0, fmaf(fin_W[1], h1, fin_b[0]));
    float f1 = fmaf(fin_W[2], h0, fmaf(fin_W[3], h1, fin_b[1]));
    hout[i] = make_float2(f0, f1);
    sqout[i] = fmaf(f0, f0, f1 * f1);
}

// ---------------------------------------------------------------------------
// Kernel 2: RBF Gram row-sums via V_WMMA_F32_16X16X4_F32.
//   A row i = [-2*h_i0, -2*h_i1, sq_i, 1]
//   B col j = [ h_j0,    h_j1,   1,   sq_j]
//   => (A*B)[i,j] = sq_i + sq_j - 2*h_i.h_j = d2(i,j)  (whole 16x16 tile per WMMA)
// Then per element: k = exp2(-log2e * max(d2,0)); acc += k * cls_W[2+j].
// One wave handles (row-tile itile, column-group cg); partial sums -> workspace.
//
// A-matrix 16x4 f32 layout (2 VGPRs): lanes 0-15 hold K=0 (v0) / K=1 (v1),
// lanes 16-31 hold K=2 (v0) / K=3 (v1).  B 4x16 mirrors this per-row.
// C/D 16x16 f32: VGPR v = row v (lanes 0-15) and row v+8 (lanes 16-31), N=lane&15.
// ---------------------------------------------------------------------------
__global__ void k_gram(const float2* __restrict__ h, const float* __restrict__ sq,
                       const float* __restrict__ clsW,
                       float* __restrict__ partials,
                       int N, int ntiles, int tiles_per_cg) {
    const int lane = threadIdx.x & 31;
    const int wid  = blockIdx.x * (blockDim.x >> 5) + (threadIdx.x >> 5);
    const int itile = wid % ntiles;
    const int cg    = wid / ntiles;
    const bool lo = (lane < 16);
    const int r  = lane & 15;

    // Build wave-constant A for this row tile
    const int gi = itile * 16 + r;
    float2 hi = h[gi];
    float sqi = sq[gi];
    v2f A;
    A.x = lo ? (-2.0f * hi.x) : sqi;
    A.y = lo ? (-2.0f * hi.y) : 1.0f;

    float acc[8];
#pragma unroll
    for (int v = 0; v < 8; ++v) acc[v] = 0.0f;

    const int jt0 = cg * tiles_per_cg;
    for (int jt = jt0; jt < jt0 + tiles_per_cg; ++jt) {
        const int jcol = jt * 16 + r;           // column this lane owns
        float2 hj = h[jcol];
        float sqj = sq[jcol];
        v2f B;
        B.x = lo ? hj.x : 1.0f;
        B.y = lo ? hj.y : sqj;

        v8f c = {};
        // D = A x B + 0  ->  16x16 tile of squared distances
        c = __builtin_amdgcn_wmma_f32_16x16x4_f32(
                /*neg_a=*/false, A, /*neg_b=*/false, B,
                /*c_mod=*/(short)0, c, /*reuse_a=*/false, /*reuse_b=*/false);

        const float wl = clsW[2 + jcol];        // classifier weight for column j
#pragma unroll
        for (int v = 0; v < 8; ++v) {
            float d2 = fmaxf(c[v], 0.0f);
            float k  = __builtin_amdgcn_exp2f(d2 * (-LOG2E));
            acc[v] = fmaf(k, wl, acc[v]);
        }
    }

    // Reduce across the 16 columns each half-wave covers (halves stay separate
    // under xor masks 8,4,2,1 -> rows v (lanes 0-15) and v+8 (lanes 16-31)).
#pragma unroll
    for (int v = 0; v < 8; ++v) {
        float a = acc[v];
        a += __shfl_xor(a, 8, 32);
        a += __shfl_xor(a, 4, 32);
        a += __shfl_xor(a, 2, 32);
        a += __shfl_xor(a, 1, 32);
        acc[v] = a;
    }

    if (r == 0) {
        const int rowoff = lo ? 0 : 8;
        float* dst = partials + (size_t)cg * N + itile * 16 + rowoff;
#pragma unroll
        for (int v = 0; v < 8; ++v) dst[v] = acc[v];
    }
}

// ---------------------------------------------------------------------------
// Kernel 3: out[i] = cls_b + cls_W[0]*h0 + cls_W[1]*h1 + sum_cg partials[cg][i]
// ---------------------------------------------------------------------------
__global__ void k_final(const float2* __restrict__ h, const float* __restrict__ partials,
                        const float* __restrict__ clsW, const float* __restrict__ clsb,
                        float* __restrict__ out, int N, int ncg) {
    int i = blockIdx.x * blockDim.x + threadIdx.x;
    if (i >= N) return;
    float2 hi = h[i];
    float s = fmaf(clsW[0], hi.x, fmaf(clsW[1], hi.y, clsb[0]));
    for (int c = 0; c < ncg; ++c) s += partials[(size_t)c * N + i];
    out[i] = s;
}

extern "C" void kernel_launch(void* const* d_in, const int* in_sizes, int n_in,
                              void* d_out, int out_size, void* d_ws, size_t ws_size,
                              hipStream_t stream) {
    const float* x      = (const float*)d_in[0];
    const float* in_W   = (const float*)d_in[1];
    const float* in_b   = (const float*)d_in[2];
    const float* in_s   = (const float*)d_in[3];
    const float* in_sh  = (const float*)d_in[4];
    const float* hid_W  = (const float*)d_in[5];
    const float* hid_b  = (const float*)d_in[6];
    const float* hid_s  = (const float*)d_in[7];
    const float* hid_sh = (const float*)d_in[8];
    const float* fin_W  = (const float*)d_in[9];
    const float* fin_b  = (const float*)d_in[10];
    const float* cls_W  = (const float*)d_in[11];
    const float* cls_b  = (const float*)d_in[12];
    float* out = (float*)d_out;

    const int N = in_sizes[0] / 2;      // 8192
    const int H = in_sizes[5] / 4;      // 4094

    // Workspace layout (floats): h[2N] | sq[N] | partials[NCG*N]
    float* ws_f     = (float*)d_ws;
    float2* h_buf   = (float2*)ws_f;                 // 2N floats
    float* sq_buf   = ws_f + 2 * (size_t)N;          // N floats
    float* partials = ws_f + 3 * (size_t)N;          // NCG*N floats

    const int NCG    = 8;               // column groups (spread waves across SIMDs)
    const int ntiles = N / 16;          // 512 row/col tiles
    const int tpcg   = ntiles / NCG;    // 64 j-tiles per wave

    // 1) sequential chain: one thread per sample
    k_chain<<<(N + 255) / 256, 256, 0, stream>>>(
        x, in_W, in_b, in_s, in_sh, hid_W, hid_b, hid_s, hid_sh,
        fin_W, fin_b, h_buf, sq_buf, H, N);

    // 2) WMMA Gram row-sums: ntiles*NCG waves, 8 waves (256 threads) per block
    const int total_waves = ntiles * NCG;            // 4096
    const int blocks = total_waves / 8;              // 512
    k_gram<<<blocks, 256, 0, stream>>>(h_buf, sq_buf, cls_W, partials, N, ntiles, tpcg);

    // 3) combine
    k_final<<<(N + 255) / 256, 256, 0, stream>>>(h_buf, partials, cls_W, cls_b, out, N, NCG);
}